// HeadDetectorLoss_49185965473872
// MI455X (gfx1250) — compile-verified
//
#include <hip/hip_runtime.h>
#include <hip/hip_bf16.h>

// ---------------------------------------------------------------------------
// HeadDetectorLoss: fused log_softmax-NLL + masked box MSE reduction.
// Memory-bound: 184.5 MB read / 23.3 TB/s  =>  ~7.9 us roofline.
// Strategy: wide NT vector loads, f32 accumulation, deterministic
// hierarchical reduction. Wave32 reductions are done on the CDNA5 matrix
// pipe with chained V_WMMA_F32_16X16X4_F32 (exact f32, fixed order).
// ---------------------------------------------------------------------------

typedef float f2  __attribute__((ext_vector_type(2)));
typedef float f4  __attribute__((ext_vector_type(4)));
typedef float v8f __attribute__((ext_vector_type(8)));

static constexpr int   kN        = 4194304;           // 2^22 samples
static constexpr int   kBlock    = 256;               // 8 waves (wave32)
static constexpr int   kGrid     = 1024;
static constexpr int   kStride   = kBlock * kGrid;    // 262144
static constexpr int   kPerThr   = kN / kStride;      // 16
static constexpr float kBoxScale = 10.0f;

// Exact full-wave32 sum using two chained f32 WMMAs.
// A layout (32-bit A, 16x4): lane L VGPR0 -> A[L%16][2*(L/16)], VGPR1 -> +1.
// With a = {v, 0} and B = all-ones: D[m][n] = v_m + v_{m+16}.
// D layout: lane L VGPR r -> D[r + 8*(L/16)][L%16]; summing the 8 VGPRs gives
// per-halfwave partials; a second identical WMMA folds the halves together.
__device__ __forceinline__ float wave32_sum_wmma(float v) {
  f2 ones; ones.x = 1.0f; ones.y = 1.0f;
  v8f c = {0.f, 0.f, 0.f, 0.f, 0.f, 0.f, 0.f, 0.f};

  f2 a; a.x = v; a.y = 0.0f;
  v8f d = __builtin_amdgcn_wmma_f32_16x16x4_f32(
      /*neg_a=*/false, a, /*neg_b=*/false, ones,
      /*c_mod=*/(short)0, c, /*reuse_a=*/false, /*reuse_b=*/false);

  float t = ((d[0] + d[1]) + (d[2] + d[3])) + ((d[4] + d[5]) + (d[6] + d[7]));

  f2 a2; a2.x = t; a2.y = 0.0f;
  v8f d2 = __builtin_amdgcn_wmma_f32_16x16x4_f32(
      false, a2, false, ones, (short)0, c, false, false);
  return d2[0];   // every element of D2 holds the full 32-lane sum
}

__global__ __launch_bounds__(kBlock)
void hdl_partial_kernel(const float* __restrict__ pred,
                        const int*   __restrict__ cls,
                        const float* __restrict__ box,
                        float* __restrict__ ws_nll,
                        float* __restrict__ ws_box,
                        float* __restrict__ ws_msk) {
  const int g = blockIdx.x * kBlock + threadIdx.x;

  const f2* __restrict__ pred2 = reinterpret_cast<const f2*>(pred); // 8B aligned (stride 24B)
  const f4* __restrict__ box4  = reinterpret_cast<const f4*>(box);  // 16B aligned

  float anll = 0.0f, abox = 0.0f, amsk = 0.0f;

#pragma unroll 4
  for (int k = 0; k < kPerThr; ++k) {
    const int i = g + k * kStride;           // coalesced: lane-consecutive samples
    const f2 p01 = __builtin_nontemporal_load(&pred2[i * 3 + 0]); // logits
    const f2 p23 = __builtin_nontemporal_load(&pred2[i * 3 + 1]); // box pred 0,1
    const f2 p45 = __builtin_nontemporal_load(&pred2[i * 3 + 2]); // box pred 2,3
    const int tc = __builtin_nontemporal_load(&cls[i]);
    const f4 tb  = __builtin_nontemporal_load(&box4[i]);

    // log-sum-exp over 2 logits: max + log(1 + exp(-|l0-l1|)); arg of log in (1,2]
    const float diff = fabsf(p01.x - p01.y);
    const float lse  = fmaxf(p01.x, p01.y) + __logf(1.0f + __expf(-diff));
    const float chosen = (tc != 0) ? p01.y : p01.x;
    anll += lse - chosen;

    const float d0 = p23.x - tb.x;
    const float d1 = p23.y - tb.y;
    const float d2 = p45.x - tb.z;
    const float d3 = p45.y - tb.w;
    const float mse = ((d0 * d0 + d1 * d1) + (d2 * d2 + d3 * d3)) * 0.25f;

    if (tc != 0) { abox += mse; amsk += 1.0f; }
  }

  // Exact wave32 reduction on the matrix pipe (EXEC is all ones here).
  const float wn = wave32_sum_wmma(anll);
  const float wb = wave32_sum_wmma(abox);
  const float wm = wave32_sum_wmma(amsk);

  __shared__ float red[kBlock / 32][3];
  const int wave = threadIdx.x >> 5;
  const int lane = threadIdx.x & 31;
  if (lane == 0) { red[wave][0] = wn; red[wave][1] = wb; red[wave][2] = wm; }
  __syncthreads();

  if (threadIdx.x == 0) {
    float sn = 0.f, sb = 0.f, sm = 0.f;
#pragma unroll
    for (int w = 0; w < kBlock / 32; ++w) {   // fixed order -> deterministic
      sn += red[w][0]; sb += red[w][1]; sm += red[w][2];
    }
    ws_nll[blockIdx.x] = sn;
    ws_box[blockIdx.x] = sb;
    ws_msk[blockIdx.x] = sm;
  }
}

// Single-wave finalize: deterministically fold kGrid block partials and emit
// the scalar loss. Uses the same WMMA wave reduction.
__global__ __launch_bounds__(32)
void hdl_finalize_kernel(const float* __restrict__ ws_nll,
                         const float* __restrict__ ws_box,
                         const float* __restrict__ ws_msk,
                         float* __restrict__ out) {
  const int lane = threadIdx.x;   // 32 threads = 1 full wave
  float anll = 0.f, abox = 0.f, amsk = 0.f;
  for (int j = lane; j < kGrid; j += 32) {    // fixed per-lane order
    anll += ws_nll[j];
    abox += ws_box[j];
    amsk += ws_msk[j];
  }
  const float tn = wave32_sum_wmma(anll);
  const float tb = wave32_sum_wmma(abox);
  const float tm = wave32_sum_wmma(amsk);

  if (lane == 0) {
    const float class_loss = tn * (1.0f / (float)kN);
    const float box_count  = 1e-6f + tm;
    out[0] = class_loss + kBoxScale * tb / box_count;
  }
}

extern "C" void kernel_launch(void* const* d_in, const int* in_sizes, int n_in,
                              void* d_out, int out_size, void* d_ws, size_t ws_size,
                              hipStream_t stream) {
  const float* pred = (const float*)d_in[0];   // (N, 6) f32
  const int*   cls  = (const int*)  d_in[1];   // (N,)   i32
  const float* box  = (const float*)d_in[2];   // (N, 4) f32
  float*       out  = (float*)d_out;

  float* ws  = (float*)d_ws;                   // 3 * kGrid floats = 12 KB
  float* wsn = ws;
  float* wsb = ws + kGrid;
  float* wsm = ws + 2 * kGrid;

  hdl_partial_kernel<<<kGrid, kBlock, 0, stream>>>(pred, cls, box, wsn, wsb, wsm);
  hdl_finalize_kernel<<<1, 32, 0, stream>>>(wsn, wsb, wsm, out);
}